// Ms_loss_38981123178915
// MI455X (gfx1250) — compile-verified
//
#include <hip/hip_runtime.h>
#include <math.h>

// Multi-similarity loss, fused 2-pass X·Xᵀ with f32 WMMA (V_WMMA_F32_16X16X4_F32).
// Never materializes the 8192x8192 sim matrix.

#define MS_ALPHA  2.0f
#define MS_BETA   5.0f
#define MS_MARGIN 0.5f
#define MS_EPS    0.2f
#define MS_N      8192
#define MS_D      128
#define MS_DK     (MS_D / 4)   // 32 k-steps of K=4

typedef __attribute__((ext_vector_type(2))) float v2f;
typedef __attribute__((ext_vector_type(8))) float v8f;

__device__ __forceinline__ v8f ms_wmma4(v2f a, v2f b, v8f c) {
  // D = A(16x4) * B(4x16) + C, all f32. 8-arg form:
  // (neg_a, A, neg_b, B, c_mod, C, reuse_a, reuse_b)
  return __builtin_amdgcn_wmma_f32_16x16x4_f32(false, a, false, b, (short)0, c,
                                               false, false);
}

// ---------------------------------------------------------------------------
// Pass 1: per-anchor hardest boundaries  min_pos[i], max_neg[i]
// One wave owns one 16-row tile and sweeps all 8192 columns -> plain stores.
// ---------------------------------------------------------------------------
__global__ __launch_bounds__(128) void ms_pass1(
    const float* __restrict__ x, const int* __restrict__ tgt2,  // int64 viewed as int pairs
    float* __restrict__ minpos, float* __restrict__ maxneg) {
  const int wave  = threadIdx.x >> 5;
  const int tid   = threadIdx.x & 31;
  const int n     = tid & 15;   // column-in-tile (B/C), row-in-tile (A)
  const int kh    = tid >> 4;   // which K pair / which M half of C
  const int rbase = (blockIdx.x * 4 + wave) * 16;

  // A panel: 16 rows x 128 k, resident in 64 VGPRs.
  const float* xr = x + (size_t)(rbase + n) * MS_D + 2 * kh;
  v2f a[MS_DK];
#pragma unroll
  for (int k = 0; k < MS_DK; ++k) a[k] = *(const v2f*)(xr + 4 * k);

  // Labels of the 8 rows this lane accumulates (C rows m = v + 8*kh).
  int trow[8];
#pragma unroll
  for (int v = 0; v < 8; ++v) trow[v] = tgt2[2 * (rbase + 8 * kh + v)];

  float mp[8], mn[8];
#pragma unroll
  for (int v = 0; v < 8; ++v) { mp[v] = INFINITY; mn[v] = -INFINITY; }

  for (int cbase = 0; cbase < MS_N; cbase += 32) {
    const float* xc0 = x + (size_t)(cbase + n) * MS_D + 2 * kh;
    const float* xc1 = xc0 + 16 * MS_D;
    __builtin_prefetch(xc0 + 32 * MS_D, 0, 0);  // next column panel -> global_prefetch_b8
    v8f c0 = {};
    v8f c1 = {};
#pragma unroll
    for (int k = 0; k < MS_DK; ++k) {          // two independent WMMA chains
      v2f b0 = *(const v2f*)(xc0 + 4 * k);
      v2f b1 = *(const v2f*)(xc1 + 4 * k);
      c0 = ms_wmma4(a[k], b0, c0);
      c1 = ms_wmma4(a[k], b1, c1);
    }
    const int tc0 = tgt2[2 * (cbase + n)];
    const int tc1 = tgt2[2 * (cbase + 16 + n)];
#pragma unroll
    for (int v = 0; v < 8; ++v) {
      float s0 = c0[v], s1 = c1[v];
      if (tc0 == trow[v]) { if (s0 < 1.0f) mp[v] = fminf(mp[v], s0); }
      else                { mn[v] = fmaxf(mn[v], s0); }
      if (tc1 == trow[v]) { if (s1 < 1.0f) mp[v] = fminf(mp[v], s1); }
      else                { mn[v] = fmaxf(mn[v], s1); }
    }
  }

  // Butterfly min/max across the 16 lanes holding one row (wave32 halves).
#pragma unroll
  for (int v = 0; v < 8; ++v) {
    float p = mp[v], q = mn[v];
    for (int m = 8; m; m >>= 1) {
      p = fminf(p, __shfl_xor(p, m, 32));
      q = fmaxf(q, __shfl_xor(q, m, 32));
    }
    mp[v] = p; mn[v] = q;
  }
  if (n == 0) {
#pragma unroll
    for (int v = 0; v < 8; ++v) {
      const int r = rbase + 8 * kh + v;
      minpos[r] = mp[v];
      maxneg[r] = mn[v];
    }
  }
}

// ---------------------------------------------------------------------------
// Pass 2: recompute sim tiles, epsilon-margin mining, weighted exp sums,
// per-anchor log1p losses, global sum+count via two atomics per half-wave.
// ---------------------------------------------------------------------------
__global__ __launch_bounds__(128) void ms_pass2(
    const float* __restrict__ x, const int* __restrict__ tgt2,
    const float* __restrict__ minpos, const float* __restrict__ maxneg,
    float* __restrict__ sumcnt) {
  const int wave  = threadIdx.x >> 5;
  const int tid   = threadIdx.x & 31;
  const int n     = tid & 15;
  const int kh    = tid >> 4;
  const int rbase = (blockIdx.x * 4 + wave) * 16;

  const float* xr = x + (size_t)(rbase + n) * MS_D + 2 * kh;
  v2f a[MS_DK];
#pragma unroll
  for (int k = 0; k < MS_DK; ++k) a[k] = *(const v2f*)(xr + 4 * k);

  int   trow[8];
  float mprow[8], mnrow[8];
#pragma unroll
  for (int v = 0; v < 8; ++v) {
    const int r = rbase + 8 * kh + v;
    trow[v]  = tgt2[2 * r];
    mprow[v] = minpos[r];
    mnrow[v] = maxneg[r];
  }

  float ps[8], ns[8];
#pragma unroll
  for (int v = 0; v < 8; ++v) { ps[v] = 0.0f; ns[v] = 0.0f; }

  for (int cbase = 0; cbase < MS_N; cbase += 32) {
    const float* xc0 = x + (size_t)(cbase + n) * MS_D + 2 * kh;
    const float* xc1 = xc0 + 16 * MS_D;
    __builtin_prefetch(xc0 + 32 * MS_D, 0, 0);
    v8f c0 = {};
    v8f c1 = {};
#pragma unroll
    for (int k = 0; k < MS_DK; ++k) {
      v2f b0 = *(const v2f*)(xc0 + 4 * k);
      v2f b1 = *(const v2f*)(xc1 + 4 * k);
      c0 = ms_wmma4(a[k], b0, c0);
      c1 = ms_wmma4(a[k], b1, c1);
    }
    const int tc0 = tgt2[2 * (cbase + n)];
    const int tc1 = tgt2[2 * (cbase + 16 + n)];
#pragma unroll
    for (int v = 0; v < 8; ++v) {
      float s0 = c0[v], s1 = c1[v];
      if (tc0 == trow[v]) {
        if (s0 < 1.0f && (s0 - MS_EPS) < mnrow[v]) ps[v] += __expf(-MS_ALPHA * (s0 - MS_MARGIN));
      } else {
        if ((s0 + MS_EPS) > mprow[v])              ns[v] += __expf( MS_BETA  * (s0 - MS_MARGIN));
      }
      if (tc1 == trow[v]) {
        if (s1 < 1.0f && (s1 - MS_EPS) < mnrow[v]) ps[v] += __expf(-MS_ALPHA * (s1 - MS_MARGIN));
      } else {
        if ((s1 + MS_EPS) > mprow[v])              ns[v] += __expf( MS_BETA  * (s1 - MS_MARGIN));
      }
    }
  }

#pragma unroll
  for (int v = 0; v < 8; ++v) {
    float p = ps[v], q = ns[v];
    for (int m = 8; m; m >>= 1) {
      p += __shfl_xor(p, m, 32);
      q += __shfl_xor(q, m, 32);
    }
    ps[v] = p; ns[v] = q;
  }

  if (n == 0) {
    float ls = 0.0f, lc = 0.0f;
#pragma unroll
    for (int v = 0; v < 8; ++v) {
      // exp() cannot underflow for s in [-1,1], so sum>0 <=> a pair was mined,
      // which also implies has_pos / has_neg.
      if (ps[v] > 0.0f && ns[v] > 0.0f) {
        ls += log1pf(ps[v]) * (1.0f / MS_ALPHA) + log1pf(ns[v]) * (1.0f / MS_BETA);
        lc += 1.0f;
      }
    }
    atomicAdd(&sumcnt[0], ls);
    atomicAdd(&sumcnt[1], lc);
  }
}

__global__ void ms_init(float* sumcnt) {
  sumcnt[0] = 0.0f;
  sumcnt[1] = 0.0f;
}

__global__ void ms_finalize(const float* __restrict__ sumcnt, float* __restrict__ out) {
  const float s = sumcnt[0], c = sumcnt[1];
  out[0] = (c > 0.0f) ? s / fmaxf(c, 1.0f) : 0.0f;
}

extern "C" void kernel_launch(void* const* d_in, const int* in_sizes, int n_in,
                              void* d_out, int out_size, void* d_ws, size_t ws_size,
                              hipStream_t stream) {
  (void)in_sizes; (void)n_in; (void)out_size; (void)ws_size;
  const float* x    = (const float*)d_in[0];
  const int*   tgt2 = (const int*)d_in[1];   // int64 labels (0..511): low dwords at even indices

  float* minpos = (float*)d_ws;              // [8192]
  float* maxneg = minpos + MS_N;             // [8192]
  float* sumcnt = maxneg + MS_N;             // [2]

  ms_init<<<1, 1, 0, stream>>>(sumcnt);
  // 512 row-tiles, 4 waves (1 tile each) per 128-thread block.
  ms_pass1<<<128, 128, 0, stream>>>(x, tgt2, minpos, maxneg);
  ms_pass2<<<128, 128, 0, stream>>>(x, tgt2, minpos, maxneg, sumcnt);
  ms_finalize<<<1, 1, 0, stream>>>(sumcnt, (float*)d_out);
}